// doc_retriever_10488310137574
// MI455X (gfx1250) — compile-verified
//
#include <hip/hip_runtime.h>
#include <hip/hip_bf16.h>
#include <math.h>

// Problem constants (match reference setup_inputs)
#define BQ 64
#define DIM 768
#define KCL 3000
#define NPER 100
#define NSEARCH 10
#define TOPK 5
#define SPARSE 128
#define NT 4            // N-subtiles (16 cols each) per wave in the WMMA GEMM

typedef float v2f __attribute__((ext_vector_type(2)));
typedef float v8f __attribute__((ext_vector_type(8)));

// ---------------------------------------------------------------------------
// Kernel 1: dist[b][c] = sum_d x[b][d]*centers[c][d] via V_WMMA_F32_16X16X4_F32
// One wave computes a 16x64 output tile (4 accumulators, A reused 4x).
// A 16x4 layout: lane L -> m=L%16, holds K = 2*(L/16) + {0,1}
// B 4x16 layout: lane L -> n=L%16, holds K = 2*(L/16) + {0,1}
// C/D layout:    lane L, vgpr r -> M = 8*(L/16)+r, N = L%16
// Out-of-range N lanes load a clamped (valid) row: their garbage only lands in
// output columns the guarded store drops, so the K-loop stays branch-free.
// ---------------------------------------------------------------------------
__global__ void dist_wmma_kernel(const float* __restrict__ x,
                                 const float* __restrict__ centers,
                                 float* __restrict__ dist) {
    const int ntile = blockIdx.x;            // 0..46 (47*64 >= 3000)
    const int mtile = blockIdx.y;            // 0..3
    const int lane  = threadIdx.x;           // wave32
    const int half  = lane >> 4;             // 0 or 1
    const int lm    = lane & 15;
    const int koff  = half * 2;

    const int m = mtile * 16 + lm;           // query row for A fragment
    const float* __restrict__ xrow = x + (size_t)m * DIM;

    const float* crow[NT];
    #pragma unroll
    for (int j = 0; j < NT; ++j) {
        int n = ntile * (16 * NT) + j * 16 + lm;
        n = (n < KCL) ? n : (KCL - 1);       // clamp: branch-free loop body
        crow[j] = centers + (size_t)n * DIM;
    }

    v8f acc[NT];
    #pragma unroll
    for (int j = 0; j < NT; ++j) acc[j] = (v8f){0.f,0.f,0.f,0.f,0.f,0.f,0.f,0.f};

    for (int k = 0; k < DIM; k += 4) {
        v2f a;
        a.x = xrow[k + koff];
        a.y = xrow[k + koff + 1];
        #pragma unroll
        for (int j = 0; j < NT; ++j) {
            v2f b;
            b.x = crow[j][k + koff];
            b.y = crow[j][k + koff + 1];
            acc[j] = __builtin_amdgcn_wmma_f32_16x16x4_f32(
                /*neg_a=*/false, a, /*neg_b=*/false, b,
                /*c_mod=*/(short)0, acc[j], /*reuse_a=*/false, /*reuse_b=*/false);
        }
    }

    const int mbase = mtile * 16 + half * 8;
    #pragma unroll
    for (int j = 0; j < NT; ++j) {
        const int ncol = ntile * (16 * NT) + j * 16 + lm;
        if (ncol < KCL) {
            #pragma unroll
            for (int r = 0; r < 8; ++r) {
                dist[(size_t)(mbase + r) * KCL + ncol] = acc[j][r];
            }
        }
    }
}

// ---------------------------------------------------------------------------
// Kernel 2: per-row top-10 of dist (destructive). One wave per query row.
// ---------------------------------------------------------------------------
__global__ void topk_clusters_kernel(float* __restrict__ dist,
                                     int* __restrict__ c_idx) {
    const int b = blockIdx.x;
    const int lane = threadIdx.x;
    float* __restrict__ row = dist + (size_t)b * KCL;

    for (int j = 0; j < NSEARCH; ++j) {
        float best = -INFINITY;
        int   bidx = 0x7FFFFFFF;
        for (int i = lane; i < KCL; i += 32) {
            float val = row[i];
            if (val > best || (val == best && i < bidx)) { best = val; bidx = i; }
        }
        #pragma unroll
        for (int off = 16; off > 0; off >>= 1) {
            float oval = __shfl_xor(best, off, 32);
            int   oidx = __shfl_xor(bidx, off, 32);
            if (oval > best || (oval == best && oidx < bidx)) { best = oval; bidx = oidx; }
        }
        // all lanes now agree on (best, bidx)
        if (lane == 0) c_idx[b * NSEARCH + j] = bidx;
        if (lane == (bidx & 31)) row[bidx] = -INFINITY;
    }
}

// ---------------------------------------------------------------------------
// Kernel 3: sparsify query: thr = 128th largest |x[b]|; v = (|x|>=thr) ? x : 0
// Exact k-th order statistic via binary search on the uint bit pattern of |x|
// (monotone for non-negative floats). One wave per row.
// ---------------------------------------------------------------------------
__global__ void sparsify_kernel(const float* __restrict__ x,
                                float* __restrict__ v) {
    const int b = blockIdx.x;
    const int lane = threadIdx.x;
    const float* __restrict__ xr = x + (size_t)b * DIM;

    unsigned lo = 0u;            // count(bits >= lo) >= SPARSE holds
    unsigned hi = 0x7F800000u;   // +inf bits: count == 0 < SPARSE
    for (int it = 0; it < 31; ++it) {
        unsigned mid = lo + ((hi - lo) >> 1);
        if (mid == lo) break;
        int cnt = 0;
        for (int i = lane; i < DIM; i += 32) {
            unsigned bits = __float_as_uint(xr[i]) & 0x7FFFFFFFu;
            cnt += (bits >= mid) ? 1 : 0;
        }
        #pragma unroll
        for (int off = 16; off > 0; off >>= 1) cnt += __shfl_xor(cnt, off, 32);
        if (cnt >= SPARSE) lo = mid; else hi = mid;
    }
    const unsigned thr = lo;     // exact bits of the 128th-largest |x|
    for (int i = lane; i < DIM; i += 32) {
        float val = xr[i];
        unsigned bits = __float_as_uint(val) & 0x7FFFFFFFu;
        v[(size_t)b * DIM + i] = (bits >= thr) ? val : 0.0f;
    }
}

// ---------------------------------------------------------------------------
// Kernel 4: scores[b][s*100+n] = v[b] . cluster_data[c_idx[b][s]][n]
// Bandwidth-dominant stage (~196 MB gathered). One wave per (b,doc) dot,
// coalesced float4 row loads (32 lanes x 16B = 512B per access).
// ---------------------------------------------------------------------------
__global__ void score_kernel(const float* __restrict__ v,
                             const float* __restrict__ cluster_data,
                             const int* __restrict__ c_idx,
                             float* __restrict__ scores) {
    const int gw   = (int)((blockIdx.x * blockDim.x + threadIdx.x) >> 5);
    const int lane = threadIdx.x & 31;
    if (gw >= BQ * NSEARCH * NPER) return;

    const int b = gw / (NSEARCH * NPER);
    const int t = gw % (NSEARCH * NPER);
    const int s = t / NPER;
    const int n = t % NPER;
    const int cl = c_idx[b * NSEARCH + s];

    const float* __restrict__ row = cluster_data + ((size_t)cl * NPER + n) * DIM;
    const float* __restrict__ vr  = v + (size_t)b * DIM;

    float acc = 0.f;
    #pragma unroll
    for (int i = 0; i < DIM / 128; ++i) {          // 6 iterations
        const int d = i * 128 + lane * 4;
        const float4 rv = *(const float4*)(row + d);
        const float4 vv = *(const float4*)(vr + d);
        acc += rv.x * vv.x + rv.y * vv.y + rv.z * vv.z + rv.w * vv.w;
    }
    #pragma unroll
    for (int off = 16; off > 0; off >>= 1) acc += __shfl_xor(acc, off, 32);
    if (lane == 0) scores[(size_t)b * (NSEARCH * NPER) + t] = acc;
}

// ---------------------------------------------------------------------------
// Kernel 5: per-row top-5 of scores (destructive), emit ret_idx (as lossless
// float values) + ret_emb gathers. One wave per query row.
// Output layout: [0,320) = ret_idx floats; [320, 320+64*5*768) = ret_emb.
// ---------------------------------------------------------------------------
__global__ void final_topk_kernel(const float* __restrict__ cluster_data,
                                  const int* __restrict__ cluster_idx,
                                  const int* __restrict__ c_idx,
                                  float* __restrict__ scores,
                                  float* __restrict__ out) {
    const int b = blockIdx.x;
    const int lane = threadIdx.x;
    float* __restrict__ srow = scores + (size_t)b * (NSEARCH * NPER);
    float* __restrict__ out_idx = out;                       // BQ*TOPK floats
    float* __restrict__ out_emb = out + BQ * TOPK;           // BQ*TOPK*DIM floats

    for (int j = 0; j < TOPK; ++j) {
        float best = -INFINITY;
        int   bidx = 0x7FFFFFFF;
        for (int i = lane; i < NSEARCH * NPER; i += 32) {
            float val = srow[i];
            if (val > best || (val == best && i < bidx)) { best = val; bidx = i; }
        }
        #pragma unroll
        for (int off = 16; off > 0; off >>= 1) {
            float oval = __shfl_xor(best, off, 32);
            int   oidx = __shfl_xor(bidx, off, 32);
            if (oval > best || (oval == best && oidx < bidx)) { best = oval; bidx = oidx; }
        }
        if (lane == (bidx & 31)) srow[bidx] = -INFINITY;

        const int s = bidx / NPER;
        const int n = bidx % NPER;
        const int cl = c_idx[b * NSEARCH + s];
        if (lane == 0) {
            // doc ids < 300000 < 2^24: float conversion is exact
            out_idx[b * TOPK + j] = (float)cluster_idx[(size_t)cl * NPER + n];
        }
        const float* __restrict__ emb = cluster_data + ((size_t)cl * NPER + n) * DIM;
        float* __restrict__ dst = out_emb + ((size_t)(b * TOPK + j)) * DIM;
        for (int d = lane; d < DIM; d += 32) dst[d] = emb[d];
    }
}

// ---------------------------------------------------------------------------
// Launch
// ---------------------------------------------------------------------------
extern "C" void kernel_launch(void* const* d_in, const int* in_sizes, int n_in,
                              void* d_out, int out_size, void* d_ws, size_t ws_size,
                              hipStream_t stream) {
    (void)in_sizes; (void)n_in; (void)out_size; (void)ws_size;

    const float* x            = (const float*)d_in[0];  // 64*768
    const float* centers      = (const float*)d_in[1];  // 3000*768
    const float* cluster_data = (const float*)d_in[2];  // 3000*100*768
    const int*   cluster_idx  = (const int*)d_in[3];    // 3000*100
    // d_in[4..6] = k(5), num_search(10), sparse_dim(128) : compile-time consts

    char* ws = (char*)d_ws;
    float* dist   = (float*)(ws);                       // 64*3000 f32 = 768000 B
    int*   c_idx  = (int*)  (ws + 768000);              // 64*10  i32 =   2560 B
    float* v      = (float*)(ws + 770560);              // 64*768 f32 = 196608 B
    float* scores = (float*)(ws + 967168);              // 64*1000 f32 = 256000 B

    // 1) fp32 WMMA GEMM: dist = x @ centers^T (16x64 tile per wave)
    dim3 g1((KCL + 16 * NT - 1) / (16 * NT), BQ / 16);  // 47 x 4 tiles
    dist_wmma_kernel<<<g1, 32, 0, stream>>>(x, centers, dist);

    // 2) top-10 clusters per query
    topk_clusters_kernel<<<BQ, 32, 0, stream>>>(dist, c_idx);

    // 3) sparsify queries (exact top-128 threshold)
    sparsify_kernel<<<BQ, 32, 0, stream>>>(x, v);

    // 4) candidate scores (bandwidth-bound gather+dot)
    const int nwaves = BQ * NSEARCH * NPER;             // 64000
    score_kernel<<<(nwaves * 32 + 255) / 256, 256, 0, stream>>>(v, cluster_data, c_idx, scores);

    // 5) final top-5 + output gathers
    final_topk_kernel<<<BQ, 32, 0, stream>>>(cluster_data, cluster_idx, c_idx,
                                             scores, (float*)d_out);
}